// Local_Mixing_64836826300529
// MI455X (gfx1250) — compile-verified
//
#include <hip/hip_runtime.h>
#include <hip/hip_bf16.h>
#include <math.h>

typedef __attribute__((ext_vector_type(2))) float v2f;
typedef __attribute__((ext_vector_type(8))) float v8f;

#define B_   8
#define C_   256
#define N_   2048
#define EPSC (256.0f * 1e-6f)   // eps added per-channel before the sum

#define RT       128            // rows of S per workgroup (8 waves x 16)
#define MT       64             // columns of S staged per chunk
#define A_STRIDE 260            // 256 + 4 pad  (4*row banks, halves disjoint)
#define B_STRIDE 80             // 64 + 16 pad  (80%64==16: lane-halves on disjoint banks)

// ---- gfx1250 async global->LDS path (types per hipcc diagnostics) --------
#if defined(__gfx1250__) && __has_builtin(__builtin_amdgcn_global_load_async_to_lds_b32) && \
    __has_builtin(__builtin_amdgcn_global_load_async_to_lds_b128) &&                        \
    __has_builtin(__builtin_amdgcn_s_wait_asynccnt)
#define HAVE_ASYNC_LDS 1
typedef int v4i32 __attribute__((vector_size(16)));
typedef __attribute__((address_space(1))) int*   g_i32p;   // printed as "__device__ int *"
typedef __attribute__((address_space(1))) v4i32* g_v4ip;
typedef __attribute__((address_space(3))) int*   l_i32p;
typedef __attribute__((address_space(3))) v4i32* l_v4ip;
__device__ __forceinline__ void async_cp_b32(const float* g, float* l) {
    __builtin_amdgcn_global_load_async_to_lds_b32((g_i32p)g, (l_i32p)l, 0, 0);
}
__device__ __forceinline__ void async_cp_b128(const float* g, float* l) {
    __builtin_amdgcn_global_load_async_to_lds_b128((g_v4ip)g, (l_v4ip)l, 0, 0);
}
#else
#define HAVE_ASYNC_LDS 0
#endif

__device__ __forceinline__ void lds_arrival_barrier() {
#if HAVE_ASYNC_LDS
    __builtin_amdgcn_s_wait_asynccnt(0);   // s_wait_asynccnt 0
#endif
    __syncthreads();
}

// stage one 256x64 B chunk: dst[c*B_STRIDE + col] = src_row0[c*N_ + col]
__device__ __forceinline__ void stage_B(float* __restrict__ dst,
                                        const float* __restrict__ src_row0, int tid) {
    const int col4 = (tid & 15) * 4;
#if HAVE_ASYNC_LDS
    #pragma unroll
    for (int p = 0; p < 16; ++p) {
        int c = p * 16 + (tid >> 4);
        async_cp_b128(src_row0 + (size_t)c * N_ + col4, dst + c * B_STRIDE + col4);
    }
#else
    #pragma unroll
    for (int p = 0; p < 16; ++p) {
        int c = p * 16 + (tid >> 4);
        const float4 v = *(const float4*)(src_row0 + (size_t)c * N_ + col4);
        *(float4*)(dst + c * B_STRIDE + col4) = v;
    }
#endif
}

// ---------------- kernel 1: inverse norms --------------------------------
__global__ void __launch_bounds__(256)
norms_kernel(const float* __restrict__ x, float* __restrict__ inv_norms) {
    int gid = blockIdx.x * blockDim.x + threadIdx.x;   // over B*N
    int b = gid / N_;
    int n = gid % N_;
    const float* xp = x + (size_t)b * C_ * N_ + n;
    float s = 0.f;
    #pragma unroll 8
    for (int c = 0; c < C_; ++c) {
        float v = xp[(size_t)c * N_];                  // coalesced across lanes
        s += v * v;
    }
    inv_norms[gid] = 1.0f / sqrtf(s + EPSC);
}

// ---------------- kernel 2: fused Gram + cosine + diag-zero + argmax -----
// S = X^T X streamed in 128x64 chunks (double-buffered B via async-to-LDS).
__global__ void __launch_bounds__(256)
argmax_sim_kernel(const float* __restrict__ x, const float* __restrict__ inv_norms,
                  int* __restrict__ idx_out) {
    extern __shared__ float smem[];
    float* A_lds = smem;                          // [RT][A_STRIDE]
    float* Bt0   = A_lds + RT * A_STRIDE;         // [C_][B_STRIDE]
    float* Bt1   = Bt0 + C_ * B_STRIDE;           // [C_][B_STRIDE]
    float* cinv  = Bt1 + C_ * B_STRIDE;           // [N_]

    const int b    = blockIdx.x / (N_ / RT);
    const int n0   = (blockIdx.x % (N_ / RT)) * RT;
    const int tid  = threadIdx.x;
    const int lane = tid & 31;
    const int wave = tid >> 5;
    const int half = lane >> 4;                   // C/D: lanes16-31 hold M=8..15
    const int l16  = lane & 15;

    const float* xb = x + (size_t)b * C_ * N_;
    const float* nv = inv_norms + (size_t)b * N_;

    // column inv-norms for the whole batch row (8 KB)
    for (int i = tid; i < N_; i += 256) cinv[i] = nv[i];

    // stage A panel transposed: A_lds[row][k] = x[b][k][n0+row]
    for (int i = 0; i < C_ / 2; ++i) {
        int c   = i * 2 + (tid >> 7);
        int col = tid & 127;
#if HAVE_ASYNC_LDS
        async_cp_b32(xb + (size_t)c * N_ + n0 + col, A_lds + col * A_STRIDE + c);
#else
        A_lds[col * A_STRIDE + c] = xb[(size_t)c * N_ + n0 + col];
#endif
    }
    stage_B(Bt0, xb, tid);                        // chunk 0
    lds_arrival_barrier();

    // per-lane state: 8 accumulator rows (M = wave*16 + half*8 + r)
    const int grow0 = n0 + wave * 16 + half * 8;
    float rowInv[8], bestV[8];
    int   bestI[8];
    #pragma unroll
    for (int r = 0; r < 8; ++r) {
        rowInv[r] = nv[grow0 + r];
        bestV[r]  = -INFINITY;
        bestI[r]  = 0;
    }

    const float* arow = &A_lds[(wave * 16 + l16) * A_STRIDE];

    for (int ic = 0; ic < N_ / MT; ++ic) {
        const int    m0 = ic * MT;
        const float* Bc = (ic & 1) ? Bt1 : Bt0;
        if (m0 + MT < N_)                          // async-prefetch next chunk
            stage_B((ic & 1) ? Bt0 : Bt1, xb + m0 + MT, tid);

        // 16 rows x 64 cols: 4 live accumulators, A fragment loaded once per k-step
        v8f acc[4];
        #pragma unroll
        for (int s = 0; s < 4; ++s) acc[s] = (v8f){0.f,0.f,0.f,0.f,0.f,0.f,0.f,0.f};

        #pragma unroll 2
        for (int ks = 0; ks < C_ / 4; ++ks) {
            const int k0 = ks * 4;
            // A 16x4 (ISA 7.12.2): lane M=l16, VGPR pair = K = {k0+2*half, +1}
            v2f a = *(const v2f*)(arow + k0 + 2 * half);
            const float* bk = Bc + (k0 + half) * B_STRIDE + l16;
            #pragma unroll
            for (int s = 0; s < 4; ++s) {
                v2f bb;
                bb.x = bk[s * 16];                 // K = k0+half
                bb.y = bk[2 * B_STRIDE + s * 16];  // K = k0+2+half
                acc[s] = __builtin_amdgcn_wmma_f32_16x16x4_f32(
                            false, a, false, bb, (short)0, acc[s], false, false);
            }
        }

        #pragma unroll
        for (int s = 0; s < 4; ++s) {
            const int   gcol = m0 + s * 16 + l16;
            const float cIv  = cinv[gcol];
            #pragma unroll
            for (int r = 0; r < 8; ++r) {
                float sv = acc[s][r] * rowInv[r] * cIv;   // cosine similarity
                if (gcol == grow0 + r) sv = 0.0f;         // fill_diagonal_(0)
                if (sv > bestV[r]) { bestV[r] = sv; bestI[r] = gcol; }
            }
        }
        lds_arrival_barrier();                     // next chunk resident / safe to swap
    }

    // cross-lane argmax within each 16-lane half (columns -> one winner per row)
    #pragma unroll
    for (int r = 0; r < 8; ++r) {
        float v = bestV[r]; int j = bestI[r];
        #pragma unroll
        for (int m = 8; m >= 1; m >>= 1) {
            float ov = __shfl_xor(v, m, 32);
            int   oj = __shfl_xor(j, m, 32);
            if (ov > v || (ov == v && oj < j)) { v = ov; j = oj; }  // first-max ties
        }
        bestV[r] = v; bestI[r] = j;
    }
    if (l16 == 0) {
        #pragma unroll
        for (int r = 0; r < 8; ++r)
            idx_out[(size_t)b * N_ + grow0 + r] = bestI[r];
    }
}

// ---------------- kernel 3: gather + gated fusion ------------------------
__global__ void __launch_bounds__(256)
fuse_kernel(const float* __restrict__ x, const float* __restrict__ W,
            const int* __restrict__ idx, float* __restrict__ out) {
    __shared__ float Ws[2 * 2 * C_];                    // [2][512]
    for (int i = threadIdx.x; i < 2 * 2 * C_; i += 256) Ws[i] = W[i];
    __syncthreads();

    int gid = blockIdx.x * blockDim.x + threadIdx.x;    // over B*N
    int b = gid / N_;
    int n = gid % N_;
    int j = idx[gid];
    const float* xb = x + (size_t)b * C_ * N_;

    float l0 = 0.f, l1 = 0.f;
    #pragma unroll 4
    for (int c = 0; c < C_; ++c) {
        float xa = xb[(size_t)c * N_ + n];              // coalesced
        float xr = xb[(size_t)c * N_ + j];              // gather (L2-resident)
        l0 += Ws[c] * xa + Ws[C_ + c] * xr;
        l1 += Ws[2 * C_ + c] * xa + Ws[3 * C_ + c] * xr;
    }
    float mx = fmaxf(l0, l1);
    float e0 = __expf(l0 - mx), e1 = __expf(l1 - mx);
    float w0 = e0 / (e0 + e1), w1 = e1 / (e0 + e1);

    float* out0 = out;                                  // out   [B,C,N]
    float* out1 = out + (size_t)B_ * C_ * N_;           // f_re  [B,C,N]
    #pragma unroll 4
    for (int c = 0; c < C_; ++c) {
        float xa = xb[(size_t)c * N_ + n];
        float xr = xb[(size_t)c * N_ + j];
        size_t o = (size_t)b * C_ * N_ + (size_t)c * N_ + n;
        out0[o] = xa * w0 + xr * w1;
        out1[o] = xr;
    }
}

// ---------------- launcher ----------------------------------------------
extern "C" void kernel_launch(void* const* d_in, const int* in_sizes, int n_in,
                              void* d_out, int out_size, void* d_ws, size_t ws_size,
                              hipStream_t stream) {
    const float* x = (const float*)d_in[0];
    const float* W = (const float*)d_in[1];
    float* out = (float*)d_out;

    float* inv_norms = (float*)d_ws;                                   // B*N f32
    int*   idx       = (int*)((char*)d_ws + (size_t)B_ * N_ * sizeof(float));

    norms_kernel<<<(B_ * N_) / 256, 256, 0, stream>>>(x, inv_norms);

    const size_t lds_bytes =
        (size_t)(RT * A_STRIDE + 2 * C_ * B_STRIDE + N_) * sizeof(float); // ~305 KB of 320 KB WGP LDS
    (void)hipFuncSetAttribute(reinterpret_cast<const void*>(argmax_sim_kernel),
                              hipFuncAttributeMaxDynamicSharedMemorySize,
                              (int)lds_bytes);
    argmax_sim_kernel<<<B_ * (N_ / RT), 256, lds_bytes, stream>>>(x, inv_norms, idx);

    fuse_kernel<<<(B_ * N_) / 256, 256, 0, stream>>>(x, W, idx, out);
}